// ZoneD_88871463289415
// MI455X (gfx1250) — compile-verified
//
#include <hip/hip_runtime.h>
#include <hip/hip_bf16.h>
#include <math.h>

// ---------------------------------------------------------------------------
// ZoneD forward for MI455X (gfx1250).  fp32 end-to-end.
// GEMMs: fp32 WMMA (v_wmma_f32_16x16x4_f32), 128x64 block macro-tile,
// 32x64 per wave, K chunked x32 and double-buffered in LDS via the CDNA5
// async path (global_load_async_to_lds_b128 + s_wait_asynccnt).
// pscan recurrences restructured as block-parallel prefix sums.
// ---------------------------------------------------------------------------

typedef __attribute__((ext_vector_type(2))) float v2f;
typedef __attribute__((ext_vector_type(8))) float v8f;

static constexpr int Bz = 4;     // batch
static constexpr int Lz = 2048;  // sequence
static constexpr int Mz = 256;   // concepts
static constexpr int Dz = 1024;  // model dim
static constexpr int Oz = 256;   // inner dim (do)

__device__ __forceinline__ float sigm(float x) { return 1.f / (1.f + expf(-x)); }

// ---------------------------------------------------------------------------
// p_at_bounds = clip(bprobs[b, bidx[b,m]], 0.1, inf);  bucket via searchsorted
// ---------------------------------------------------------------------------
__global__ void bounds_kernel(const float* __restrict__ bprobs,
                              const int* __restrict__ bidx,
                              float* __restrict__ p_at,
                              int* __restrict__ bucket)
{
    int t = blockIdx.x * blockDim.x + threadIdx.x;
    if (t < Bz * Mz) {
        int b = t / Mz;
        p_at[t] = fmaxf(bprobs[b * Lz + bidx[t]], 0.1f);
    }
    if (t < Bz * Lz) {
        int b = t / Lz, l = t % Lz;
        const int* bi = bidx + b * Mz;
        int lo = 0, hi = Mz;                 // searchsorted side='right'
        while (lo < hi) { int mid = (lo + hi) >> 1; if (bi[mid] <= l) lo = mid + 1; else hi = mid; }
        bucket[t] = max(lo - 1, 0);
    }
}

// ---------------------------------------------------------------------------
// Concept-level pscan with h0.  a depends only on (b,step): precompute
// A, invA (and p) once per block in LDS, then stream 256 channels/block.
// ---------------------------------------------------------------------------
__global__ void concept_scan_kernel(const float* __restrict__ concept,
                                    const float* __restrict__ p_at,
                                    float* __restrict__ smoothed)
{
    const int blocksPerB = Dz / 256;
    int b  = blockIdx.x / blocksPerB;
    int ch = (blockIdx.x % blocksPerB) * 256 + threadIdx.x;

    __shared__ float sA[Mz - 1], sInvA[Mz - 1], sP[Mz - 1];
    if (threadIdx.x == 0) {
        float P = 0.f;
        for (int j = 0; j < Mz - 1; ++j) {
            float p   = p_at[b * Mz + 1 + j];
            float dec = fmaxf(1.f - p, 1e-7f);
            P += logf(dec);
            float logA = fminf(fmaxf(P, -80.f), 0.f);
            sA[j]    = expf(logA);
            sInvA[j] = expf(-logA);
            sP[j]    = p;
        }
    }
    __syncthreads();

    float h0 = concept[((size_t)b * Mz) * Dz + ch];
    smoothed[((size_t)b * Mz) * Dz + ch] = h0;
    float S = 0.f;
    for (int j = 0; j < Mz - 1; ++j) {
        float bv = sP[j] * concept[((size_t)b * Mz + 1 + j) * Dz + ch];
        S += bv * sInvA[j];
        smoothed[((size_t)b * Mz + 1 + j) * Dz + ch] = sA[j] * (S + h0);
    }
}

// ---------------------------------------------------------------------------
// fp32 WMMA GEMM, LDS-tiled:  C = epi(A(MxK) @ W(NxK)^T)
//   block: 128 threads (4 waves), macro tile 128x64, wave tile 32x64
//   K chunked by 32, double-buffered in LDS, filled with
//   global_load_async_to_lds_b128 (ASYNCcnt) -> s_wait_asynccnt -> barrier.
//   LDS row stride 36 floats: 16B-aligned rows + conflict-free fragment reads.
// EPI: 0 = none, 1 = sigmoid(x + bias[n])
// ---------------------------------------------------------------------------
template <int EPI>
__global__ __launch_bounds__(128)
void gemm_wmma_tiled(const float* __restrict__ A, const float* __restrict__ W,
                     const float* __restrict__ bias, float* __restrict__ C,
                     int M, int N, int K)
{
    constexpr int BM = 128, BN = 64, KT = 32, LDST = 36;
    __shared__ float AT[2][BM * LDST];
    __shared__ float WT[2][BN * LDST];

    const int nbn  = N / BN;
    const int bm0  = (blockIdx.x / nbn) * BM;
    const int bn0  = (blockIdx.x % nbn) * BN;
    const int tid  = threadIdx.x;
    const int wv   = tid >> 5;
    const int lane = tid & 31;
    const int half = lane >> 4;
    const int l15  = lane & 15;

    // async-copy one K-chunk (A: 128x32, W: 64x32) into LDS buffer `buf`
    auto issue = [&](int buf, int k0) {
#pragma unroll
        for (int i = 0; i < 8; ++i) {            // A: 1024 16B chunks / 128 thr
            int idx = i * 128 + tid;
            int row = idx >> 3, kq = idx & 7;
            uint32_t lds = (uint32_t)(uintptr_t)&AT[buf][row * LDST + kq * 4];
            uint64_t g   = (uint64_t)(uintptr_t)(A + (size_t)(bm0 + row) * K + k0 + kq * 4);
            asm volatile("global_load_async_to_lds_b128 %0, %1, off"
                         :: "v"(lds), "v"(g) : "memory");
        }
#pragma unroll
        for (int i = 0; i < 4; ++i) {            // W: 512 16B chunks / 128 thr
            int idx = i * 128 + tid;
            int row = idx >> 3, kq = idx & 7;
            uint32_t lds = (uint32_t)(uintptr_t)&WT[buf][row * LDST + kq * 4];
            uint64_t g   = (uint64_t)(uintptr_t)(W + (size_t)(bn0 + row) * K + k0 + kq * 4);
            asm volatile("global_load_async_to_lds_b128 %0, %1, off"
                         :: "v"(lds), "v"(g) : "memory");
        }
    };

    v8f acc[2][4] = {};
    issue(0, 0);
    const int nch = K / KT;
    int buf = 0;
    for (int c = 0; c < nch; ++c) {
        asm volatile("s_wait_asynccnt 0x0" ::: "memory");  // my chunk-c loads done
        __syncthreads();                                   // everyone's are done
        if (c + 1 < nch) issue(buf ^ 1, (c + 1) * KT);     // prefetch next chunk

        const float* At = AT[buf];
        const float* Wt = WT[buf];
#pragma unroll
        for (int k = 0; k < KT; k += 4) {
            v2f af[2], bf[4];
#pragma unroll
            for (int mt = 0; mt < 2; ++mt) {
                int o = (wv * 32 + mt * 16 + l15) * LDST + k + 2 * half;
                af[mt].x = At[o]; af[mt].y = At[o + 1];
            }
#pragma unroll
            for (int nt = 0; nt < 4; ++nt) {
                int o = (nt * 16 + l15) * LDST + k + 2 * half;
                bf[nt].x = Wt[o]; bf[nt].y = Wt[o + 1];
            }
#pragma unroll
            for (int mt = 0; mt < 2; ++mt)
#pragma unroll
                for (int nt = 0; nt < 4; ++nt)
                    acc[mt][nt] = __builtin_amdgcn_wmma_f32_16x16x4_f32(
                        false, af[mt], false, bf[nt], (short)0, acc[mt][nt], false, false);
        }
        buf ^= 1;
    }

#pragma unroll
    for (int mt = 0; mt < 2; ++mt)
#pragma unroll
        for (int nt = 0; nt < 4; ++nt) {
            int n = bn0 + nt * 16 + l15;
            float bb = (EPI == 1) ? bias[n] : 0.f;
#pragma unroll
            for (int v = 0; v < 8; ++v) {
                int m = bm0 + wv * 32 + mt * 16 + half * 8 + v;
                float val = acc[mt][nt][v];
                if (EPI == 1) val = sigm(val + bb);
                C[(size_t)m * N + n] = val;
            }
        }
}

// ---------------------------------------------------------------------------
// h = (1 - bp)*gate*enc + smoothed[b, bucket]   (in-place over gate buffer)
// ---------------------------------------------------------------------------
__global__ void combine_kernel(const float* __restrict__ enc,
                               const float* __restrict__ bp,
                               const int* __restrict__ bucket,
                               const float* __restrict__ smoothed,
                               float* __restrict__ h)
{
    size_t idx = (size_t)blockIdx.x * blockDim.x + threadIdx.x;  // B*L*D
    int n = (int)(idx % Dz);
    size_t ml = idx / Dz;            // b*L + l
    int b = (int)(ml / Lz);
    float g = h[idx];
    h[idx] = (1.f - bp[ml]) * g * enc[idx] +
             smoothed[((size_t)b * Mz + bucket[ml]) * Dz + n];
}

// ---------------------------------------------------------------------------
// Depthwise causal conv, 4 taps, left pad 3.  x,xc: (B,L,O)
// ---------------------------------------------------------------------------
__global__ void conv_kernel(const float* __restrict__ x,
                            const float* __restrict__ cw,   // (O,4)
                            const float* __restrict__ cb,   // (O)
                            float* __restrict__ xc)
{
    size_t idx = (size_t)blockIdx.x * blockDim.x + threadIdx.x;  // B*L*O
    int c = (int)(idx % Oz);
    size_t bl = idx / Oz;
    int l = (int)(bl % Lz);
    size_t bbase = (bl - l) * Oz;     // batch start
    float acc = cb[c];
#pragma unroll
    for (int k = 0; k < 4; ++k) {
        int ls = l - 3 + k;
        if (ls >= 0) acc += x[bbase + (size_t)ls * Oz + c] * cw[c * 4 + k];
    }
    xc[idx] = acc;
}

// ---------------------------------------------------------------------------
// a_t = sigmoid(log_a)^(8r);  b_t = sqrt(clip(1-a_t^2,1e-6))*(i*xc)
// written transposed to (B, O, L) for contiguous prefix scans.
// ---------------------------------------------------------------------------
__global__ void ab_kernel(const float* __restrict__ r, const float* __restrict__ i,
                          const float* __restrict__ xc, const float* __restrict__ log_a,
                          float* __restrict__ aT, float* __restrict__ bT)
{
    size_t idx = (size_t)blockIdx.x * blockDim.x + threadIdx.x;  // B*L*O
    int c = (int)(idx % Oz);
    size_t bl = idx / Oz;
    int l = (int)(bl % Lz);
    int b = (int)(bl / Lz);
    float ab = sigm(log_a[c]);
    float at = expf(8.f * r[idx] * logf(ab));
    float bt = sqrtf(fmaxf(1.f - at * at, 1e-6f)) * (i[idx] * xc[idx]);
    size_t tp = ((size_t)(b * Oz + c)) * Lz + l;
    aT[tp] = at; bT[tp] = bt;
}

// ---------------------------------------------------------------------------
// pscan over L via two block-parallel prefix sums (reference formulation):
//   logA = clip(cumsum(log(clip(a,1e-7))), -80, 0)
//   h    = exp(logA) * cumsum(b * exp(-logA))
// one block of 256 threads per (b, channel); 8 elements / thread.
// ---------------------------------------------------------------------------
__global__ __launch_bounds__(256)
void pscan_kernel(const float* __restrict__ aT, const float* __restrict__ bT,
                  float* __restrict__ h)
{
    constexpr int T = 256, E = Lz / T;
    int bc = blockIdx.x;                 // b*O + c
    int b = bc / Oz, c = bc % Oz;
    int tid = threadIdx.x;
    const float* ap = aT + (size_t)bc * Lz + tid * E;
    const float* bp = bT + (size_t)bc * Lz + tid * E;

    __shared__ float sb[T];

    float la[E]; float s = 0.f;
#pragma unroll
    for (int j = 0; j < E; ++j) { la[j] = logf(fmaxf(ap[j], 1e-7f)); s += la[j]; }

    sb[tid] = s; __syncthreads();
#pragma unroll
    for (int o = 1; o < T; o <<= 1) {
        float t = (tid >= o) ? sb[tid - o] : 0.f;
        __syncthreads(); sb[tid] += t; __syncthreads();
    }
    float run = sb[tid] - s;             // exclusive prefix
    __syncthreads();

    float logA[E];
#pragma unroll
    for (int j = 0; j < E; ++j) { run += la[j]; logA[j] = fminf(fmaxf(run, -80.f), 0.f); }

    float cv[E]; float s2 = 0.f;
#pragma unroll
    for (int j = 0; j < E; ++j) { cv[j] = bp[j] * expf(-logA[j]); s2 += cv[j]; }

    sb[tid] = s2; __syncthreads();
#pragma unroll
    for (int o = 1; o < T; o <<= 1) {
        float t = (tid >= o) ? sb[tid - o] : 0.f;
        __syncthreads(); sb[tid] += t; __syncthreads();
    }
    run = sb[tid] - s2;

    float* hp = h + ((size_t)b * Lz + tid * E) * Oz + c;
#pragma unroll
    for (int j = 0; j < E; ++j) { run += cv[j]; hp[(size_t)j * Oz] = expf(logA[j]) * run; }
}

// ---------------------------------------------------------------------------
// rmsnorm per row (row length Dn), 256 threads / row
// ---------------------------------------------------------------------------
__global__ __launch_bounds__(256)
void rmsnorm_kernel(const float* __restrict__ X, const float* __restrict__ w,
                    float* __restrict__ Y, int Dn)
{
    int row = blockIdx.x;
    const float* x = X + (size_t)row * Dn;
    float* y = Y + (size_t)row * Dn;
    float s = 0.f;
    for (int j = threadIdx.x; j < Dn; j += 256) { float v = x[j]; s += v * v; }
    __shared__ float red[256];
    red[threadIdx.x] = s; __syncthreads();
    for (int o = 128; o > 0; o >>= 1) {
        if (threadIdx.x < o) red[threadIdx.x] += red[threadIdx.x + o];
        __syncthreads();
    }
    float scale = rsqrtf(red[0] / (float)Dn + 1e-6f);
    for (int j = threadIdx.x; j < Dn; j += 256) y[j] = x[j] * scale * w[j];
}

// ---------------------------------------------------------------------------
extern "C" void kernel_launch(void* const* d_in, const int* in_sizes, int n_in,
                              void* d_out, int out_size, void* d_ws, size_t ws_size,
                              hipStream_t stream)
{
    (void)in_sizes; (void)n_in; (void)out_size; (void)ws_size;

    const float* concept  = (const float*)d_in[0];
    const float* enc      = (const float*)d_in[1];
    const float* bprobs   = (const float*)d_in[2];
    const int*   bidx     = (const int*)  d_in[3];
    const float* gate_W   = (const float*)d_in[4];
    const float* gate_b   = (const float*)d_in[5];
    const float* down_W   = (const float*)d_in[6];
    const float* up_W     = (const float*)d_in[7];
    const float* norm_out = (const float*)d_in[8];
    const float* conv_w   = (const float*)d_in[9];
    const float* conv_b   = (const float*)d_in[10];
    const float* Wr_w     = (const float*)d_in[11];
    const float* Wr_b     = (const float*)d_in[12];
    const float* Wi_w     = (const float*)d_in[13];
    const float* Wi_b     = (const float*)d_in[14];
    const float* log_a    = (const float*)d_in[15];
    const float* out_w    = (const float*)d_in[16];
    const float* norm_w   = (const float*)d_in[17];
    float* out = (float*)d_out;

    // ---- workspace carve-out (fp32 elements) ----
    const size_t nBLD = (size_t)Bz * Lz * Dz;   // 8,388,608
    const size_t nBLO = (size_t)Bz * Lz * Oz;   // 2,097,152
    char* wsp = (char*)d_ws;
    auto takeF = [&](size_t nfloats) { float* p = (float*)wsp; wsp += nfloats * sizeof(float); return p; };
    float* smoothed = takeF((size_t)Bz * Mz * Dz);
    float* p_at     = takeF((size_t)Bz * Mz);
    int*   bucket   = (int*)wsp; wsp += (size_t)Bz * Lz * sizeof(int);
    float* hbig     = takeF(nBLD);   // gate/combine output, reused for up output
    float* h_io     = takeF(nBLO);   // rec-layer input/output
    float* xc       = takeF(nBLO);
    float* rbuf     = takeF(nBLO);
    float* ibuf     = takeF(nBLO);
    float* aT       = takeF(nBLO);
    float* bT       = takeF(nBLO);
    float* hscan    = takeF(nBLO);
    float* tmp      = takeF(nBLO);

    const int ML = Bz * Lz;            // 8192 rows in every GEMM

    auto gemm_blocks = [](int M, int N) { return (M / 128) * (N / 64); };

    // 1) boundaries
    bounds_kernel<<<(Bz * Lz + 255) / 256, 256, 0, stream>>>(bprobs, bidx, p_at, bucket);
    // 2) concept scan -> smoothed
    concept_scan_kernel<<<Bz * (Dz / 256), 256, 0, stream>>>(concept, p_at, smoothed);
    // 3) gate = sigmoid(enc @ gate_W^T + gate_b)
    gemm_wmma_tiled<1><<<gemm_blocks(ML, Dz), 128, 0, stream>>>(enc, gate_W, gate_b, hbig, ML, Dz, Dz);
    // 4) combine with plugback (in place on hbig)
    combine_kernel<<<(int)(nBLD / 256), 256, 0, stream>>>(enc, bprobs, bucket, smoothed, hbig);
    // 5) down projection
    gemm_wmma_tiled<0><<<gemm_blocks(ML, Oz), 128, 0, stream>>>(hbig, down_W, nullptr, h_io, ML, Oz, Dz);

    // 6) three recurrent layers
    for (int l = 0; l < 3; ++l) {
        const float* cw = conv_w + (size_t)l * Oz * 4;
        const float* cb = conv_b + (size_t)l * Oz;
        const float* wr = Wr_w + (size_t)l * Oz * Oz;
        const float* br = Wr_b + (size_t)l * Oz;
        const float* wi = Wi_w + (size_t)l * Oz * Oz;
        const float* bi = Wi_b + (size_t)l * Oz;
        const float* la = log_a + (size_t)l * Oz;
        const float* wo = out_w + (size_t)l * Oz * Oz;
        const float* nw = norm_w + (size_t)l * Oz;

        conv_kernel<<<(int)(nBLO / 256), 256, 0, stream>>>(h_io, cw, cb, xc);
        gemm_wmma_tiled<1><<<gemm_blocks(ML, Oz), 128, 0, stream>>>(xc, wr, br, rbuf, ML, Oz, Oz);
        gemm_wmma_tiled<1><<<gemm_blocks(ML, Oz), 128, 0, stream>>>(xc, wi, bi, ibuf, ML, Oz, Oz);
        ab_kernel<<<(int)(nBLO / 256), 256, 0, stream>>>(rbuf, ibuf, xc, la, aT, bT);
        pscan_kernel<<<Bz * Oz, 256, 0, stream>>>(aT, bT, hscan);
        gemm_wmma_tiled<0><<<gemm_blocks(ML, Oz), 128, 0, stream>>>(hscan, wo, nullptr, tmp, ML, Oz, Oz);
        rmsnorm_kernel<<<ML, 256, 0, stream>>>(tmp, nw, h_io, Oz);
    }

    // 7) up projection, 8) final rmsnorm -> out
    gemm_wmma_tiled<0><<<gemm_blocks(ML, Dz), 128, 0, stream>>>(h_io, up_W, nullptr, hbig, ML, Dz, Oz);
    rmsnorm_kernel<<<ML, 256, 0, stream>>>(hbig, norm_out, out, Dz);
}